// VanillaTransformer_74139725464235
// MI455X (gfx1250) — compile-verified
//
#include <hip/hip_runtime.h>
#include <hip/hip_bf16.h>
#include <math.h>

typedef __attribute__((ext_vector_type(2))) float v2f;
typedef __attribute__((ext_vector_type(8))) float v8f;

// ---------------------------------------------------------------------------
// Embedding + deterministic position encoding
// x[t,d] = tok_emb[idx[t],d]  (+ pos terms on channels 0,1,2)
// ---------------------------------------------------------------------------
__global__ void embed_kernel(const int* __restrict__ idx,
                             const float* __restrict__ tok_emb,
                             float* __restrict__ x, int S) {
    const int D = 36;
    int t = blockIdx.x;          // token index in [0, B*S)
    int d = threadIdx.x;
    if (d >= D) return;
    int s  = t % S;
    int v  = idx[t];
    float val = tok_emb[(size_t)v * D + d];
    float p = (float)s;
    if (d == 0)      val += p;
    else if (d == 1) val += (1.0f / 0.6931471805599453f) - 1.0f / logf(p + 2.0f);
    else if (d == 2) val += p * p;
    x[(size_t)t * D + d] = val;
}

// ---------------------------------------------------------------------------
// Generic fp32 WMMA GEMM:  D[M,N] = A[M,36] * W[N,36]^T  (+ Cin if non-null)
// One 16x16 output tile per wave, K=36 done as 9 chained v_wmma_f32_16x16x4_f32.
// A-fragment (16x4 MxK): lanes 0-15 hold M=lr, {K,K+1}; lanes 16-31 hold {K+2,K+3}.
// B-fragment (4x16 KxN): lane lr holds column n=lr, same K-pair split by half.
// C/D (16x16): VGPR r -> row r (lanes 0-15) / row r+8 (lanes 16-31), col = lr.
// ---------------------------------------------------------------------------
__global__ __launch_bounds__(128) void wmma_gemm36(
        const float* __restrict__ A,
        const float* __restrict__ W,
        const float* __restrict__ Cin,   // nullable; row stride ldd
        float* __restrict__ Dout,
        int N, int ldd) {
    const int K = 36;
    int lane  = threadIdx.x & 31;
    int wave  = threadIdx.x >> 5;
    int half  = lane >> 4;      // 0: lanes 0-15, 1: lanes 16-31
    int lr    = lane & 15;
    int mtile = blockIdx.x;
    int ntile = blockIdx.y * (blockDim.x >> 5) + wave;
    if (ntile * 16 >= N) return;               // wave-uniform exit: EXEC stays all-1s

    int m = mtile * 16 + lr;                   // A row this lane loads
    int n = ntile * 16 + lr;                   // W row / output column this lane owns
    bool nvalid = (n < N);

    v2f a[9], b[9];
#pragma unroll
    for (int kk = 0; kk < 9; ++kk) {
        int k = kk * 4 + half * 2;
        a[kk].x = A[(size_t)m * K + k];
        a[kk].y = A[(size_t)m * K + k + 1];
        b[kk].x = nvalid ? W[(size_t)n * K + k]     : 0.0f;
        b[kk].y = nvalid ? W[(size_t)n * K + k + 1] : 0.0f;
    }

    v8f c = {};
    if (Cin) {
#pragma unroll
        for (int r = 0; r < 8; ++r) {
            int row = mtile * 16 + r + half * 8;
            c[r] = nvalid ? Cin[(size_t)row * ldd + n] : 0.0f;
        }
    }

#pragma unroll
    for (int kk = 0; kk < 9; ++kk) {
        // (neg_a, A, neg_b, B, c_mod, C, reuse_a, reuse_b)
        c = __builtin_amdgcn_wmma_f32_16x16x4_f32(
                false, a[kk], false, b[kk], (short)0, c, false, false);
    }

#pragma unroll
    for (int r = 0; r < 8; ++r) {
        int row = mtile * 16 + r + half * 8;
        if (nvalid) Dout[(size_t)row * ldd + n] = c[r];
    }
}

// ---------------------------------------------------------------------------
// Causal attention, head_dim = 2 (WMMA-hostile: K=2), online softmax.
// One wave per (b, h, q); lanes stride over keys; butterfly merge of
// (m, s, o0, o1) partials via __shfl_xor. qkv row layout: [q(36) k(36) v(36)].
// ---------------------------------------------------------------------------
__global__ __launch_bounds__(32) void attn_kernel(
        const float* __restrict__ qkv,
        float* __restrict__ out, int S) {
    int q = blockIdx.x;
    int h = blockIdx.y;
    int b = blockIdx.z;
    int lane = threadIdx.x;

    const float scale = 0.7071067811865476f;   // 1/sqrt(2)
    const float* rowq = qkv + (size_t)(b * S + q) * 108;
    float q0 = rowq[h * 2 + 0];
    float q1 = rowq[h * 2 + 1];

    float m = -3.0e38f, s = 0.0f, o0 = 0.0f, o1 = 0.0f;
    for (int j = lane; j <= q; j += 32) {
        const float* rj = qkv + (size_t)(b * S + j) * 108;
        float sc = (q0 * rj[36 + h * 2] + q1 * rj[36 + h * 2 + 1]) * scale;
        float v0 = rj[72 + h * 2], v1 = rj[72 + h * 2 + 1];
        float nm = fmaxf(m, sc);
        float em = __expf(m - nm);     // underflows to 0 for the sentinel, no NaN
        float es = __expf(sc - nm);
        s  = s  * em + es;
        o0 = o0 * em + es * v0;
        o1 = o1 * em + es * v1;
        m  = nm;
    }
    // butterfly merge across the wave32
    for (int off = 16; off > 0; off >>= 1) {
        float om  = __shfl_xor(m,  off, 32);
        float os  = __shfl_xor(s,  off, 32);
        float oa  = __shfl_xor(o0, off, 32);
        float ob  = __shfl_xor(o1, off, 32);
        float nm = fmaxf(m, om);
        float ca = __expf(m  - nm);
        float cb = __expf(om - nm);
        s  = s  * ca + os * cb;
        o0 = o0 * ca + oa * cb;
        o1 = o1 * ca + ob * cb;
        m  = nm;
    }
    if (lane == 0) {
        float inv = 1.0f / s;
        size_t o = (size_t)(b * S + q) * 36 + h * 2;
        out[o]     = o0 * inv;
        out[o + 1] = o1 * inv;
    }
}

// ---------------------------------------------------------------------------
// GLU nonlinearity: h[t,f] = relu(gv[t,f]) * gv[t,36+f]
// ---------------------------------------------------------------------------
__global__ void glu_kernel(const float* __restrict__ gv,
                           float* __restrict__ h, int n) {
    int i = blockIdx.x * blockDim.x + threadIdx.x;
    if (i >= n) return;
    int t = i / 36, f = i % 36;
    float g = gv[(size_t)t * 72 + f];
    float v = gv[(size_t)t * 72 + 36 + f];
    h[i] = (g > 0.0f ? g : 0.0f) * v;
}

// ---------------------------------------------------------------------------
// Host-side orchestration
// ---------------------------------------------------------------------------
extern "C" void kernel_launch(void* const* d_in, const int* in_sizes, int n_in,
                              void* d_out, int out_size, void* d_ws, size_t ws_size,
                              hipStream_t stream) {
    const int*   idx        = (const int*)  d_in[0];
    const float* tok_emb    = (const float*)d_in[1];
    const float* in_proj_w  = (const float*)d_in[2];   // [L,108,36]
    const float* out_proj_w = (const float*)d_in[3];   // [L,36,36]
    const float* ff_in_w    = (const float*)d_in[4];   // [L,72,36]
    const float* ff_out_w   = (const float*)d_in[5];   // [L,36,36]
    const float* head_w     = (const float*)d_in[6];   // [32000,36]
    float*       out        = (float*)d_out;           // [4096,32000]

    const int B = 2, S = 2048, D = 36, L = 7, F = 36, V = 32000;
    const int T = B * S;                 // 4096 tokens
    const int MT = T / 16;               // 256 M-tiles

    // workspace layout (fp32): x | qkv | attn_out | gv | hbuf  (~4.5 MB)
    float* x        = (float*)d_ws;
    float* qkv      = x        + (size_t)T * D;
    float* attn_out = qkv      + (size_t)T * 3 * D;
    float* gv       = attn_out + (size_t)T * D;
    float* hbuf     = gv       + (size_t)T * 2 * F;

    embed_kernel<<<T, 64, 0, stream>>>(idx, tok_emb, x, S);

    for (int l = 0; l < L; ++l) {
        // qkv = x @ in_proj^T : N=108 -> 7 tiles
        wmma_gemm36<<<dim3(MT, 2), 128, 0, stream>>>(
            x, in_proj_w + (size_t)l * 108 * 36, nullptr, qkv, 108, 108);

        attn_kernel<<<dim3(S, 18, B), 32, 0, stream>>>(qkv, attn_out, S);

        // x += attn_out @ out_proj^T : N=36 -> 3 tiles
        wmma_gemm36<<<dim3(MT, 1), 128, 0, stream>>>(
            attn_out, out_proj_w + (size_t)l * 36 * 36, x, x, 36, 36);

        // gv = x @ ff_in^T : N=72 -> 5 tiles
        wmma_gemm36<<<dim3(MT, 2), 128, 0, stream>>>(
            x, ff_in_w + (size_t)l * 72 * 36, nullptr, gv, 72, 72);

        glu_kernel<<<(T * F + 255) / 256, 256, 0, stream>>>(gv, hbuf, T * F);

        // x += hbuf @ ff_out^T : N=36
        wmma_gemm36<<<dim3(MT, 1), 128, 0, stream>>>(
            hbuf, ff_out_w + (size_t)l * 36 * 36, x, x, 36, 36);
    }

    // logits = x @ head_w^T : N=32000 -> 2000 tiles (store-bandwidth bound)
    wmma_gemm36<<<dim3(MT, 500), 128, 0, stream>>>(
        x, head_w, nullptr, out, V, V);
    (void)in_sizes; (void)n_in; (void)out_size; (void)ws_size;
}